// Model_77610059038836
// MI455X (gfx1250) — compile-verified
//
#include <hip/hip_runtime.h>
#include <cmath>

typedef float v2f __attribute__((ext_vector_type(2)));
typedef float v8f __attribute__((ext_vector_type(8)));

#define EPSF 1e-9f

// ---------------- 1) vertex transform + projection ----------------
__global__ void vert_kernel(const float* __restrict__ verts,
                            const float* __restrict__ Km,
                            const float* __restrict__ Rm,
                            const float* __restrict__ tm,
                            float4* __restrict__ uvz,
                            int B, int V, int S)
{
    int tid = blockIdx.x * blockDim.x + threadIdx.x;
    if (tid >= B * V) return;
    int b = tid / V;
    const float* vp = verts + (size_t)tid * 3;
    const float* Rb = Rm + b * 9;
    const float* tb = tm + b * 3;
    const float* Kb = Km + b * 9;
    float vx = Rb[0]*vp[0] + Rb[1]*vp[1] + Rb[2]*vp[2] + tb[0];
    float vy = Rb[3]*vp[0] + Rb[4]*vp[1] + Rb[5]*vp[2] + tb[1];
    float vz = Rb[6]*vp[0] + Rb[7]*vp[1] + Rb[8]*vp[2] + tb[2];
    float inv = 1.0f / (vz + EPSF);
    float xh = vx * inv, yh = vy * inv;
    float pu = Kb[0]*xh + Kb[1]*yh + Kb[2];
    float pv = Kb[3]*xh + Kb[4]*yh + Kb[5];
    float Sf = (float)S;
    float u = 2.0f * (pu - 0.5f * Sf) / Sf;
    float v = 2.0f * ((Sf - pv) - 0.5f * Sf) / Sf;
    uvz[tid] = make_float4(u, v, vz, 0.0f);
}

// ---------------- 2) per-face plane coefficients ----------------
// w1 = a1*PX + b1*PY + c1 ; w2 likewise ; invz = Az*PX + Bz*PY + Cz
// invalid/back faces get w1 == -1 (never "inside"), invz coeffs 0.
__global__ void face_kernel(const int* __restrict__ faces,
                            const float4* __restrict__ uvz,
                            float4* __restrict__ P1, float4* __restrict__ P2,
                            float4* __restrict__ PZ, float4* __restrict__ ZZ,
                            int B, int V, int F, int Fp)
{
    int tid = blockIdx.x * blockDim.x + threadIdx.x;
    if (tid >= B * Fp) return;
    int b = tid / Fp, f = tid % Fp;
    float4 o1 = make_float4(0.f, 0.f, -1.f, 0.f);
    float4 o2 = make_float4(0.f, 0.f, 0.f, 0.f);
    float4 oz = make_float4(0.f, 0.f, 0.f, 0.f);
    float4 ozz = make_float4(1.f, 1.f, 1.f, 0.f);
    if (f < F) {
        const int* fv = faces + ((size_t)b * F + f) * 3;
        float4 p0 = uvz[(size_t)b * V + fv[0]];
        float4 p1 = uvz[(size_t)b * V + fv[1]];
        float4 p2 = uvz[(size_t)b * V + fv[2]];
        float x0 = p0.x, y0 = p0.y, z0 = p0.z;
        float e1x = p1.x - x0, e1y = p1.y - y0;
        float e2x = p2.x - x0, e2y = p2.y - y0;
        float den = e1x * e2y - e2x * e1y;
        bool  valid = fabsf(den) > 1e-8f;
        float dens = valid ? den : 1.0f;
        bool  front = fminf(fminf(z0, p1.z), p2.z) > EPSF;
        ozz = make_float4(z0, p1.z, p2.z, 0.0f);
        if (valid && front) {
            float a1 =  e2y / dens, b1 = -e2x / dens, c1 = (y0*e2x - x0*e2y) / dens;
            float a2 = -e1y / dens, b2 =  e1x / dens, c2 = (x0*e1y - y0*e1x) / dens;
            float a0 = -(a1 + a2), b0 = -(b1 + b2), c0 = 1.0f - c1 - c2;
            float iz0 = 1.0f / z0, iz1 = 1.0f / p1.z, iz2 = 1.0f / p2.z;
            o1 = make_float4(a1, b1, c1, 0.f);
            o2 = make_float4(a2, b2, c2, 0.f);
            oz = make_float4(a0*iz0 + a1*iz1 + a2*iz2,
                             b0*iz0 + b1*iz1 + b2*iz2,
                             c0*iz0 + c1*iz1 + c2*iz2, 0.f);
        }
    }
    size_t o = (size_t)b * Fp + f;
    P1[o] = o1; P2[o] = o2; PZ[o] = oz; ZZ[o] = ozz;
}

// ---------------- 3) precompute tanh(textures) ----------------
__global__ void tanh_kernel(const float* __restrict__ tex, float* __restrict__ TT,
                            int nValid, int nTot)
{
    int i = blockIdx.x * blockDim.x + threadIdx.x;
    if (i < nTot) TT[i] = (i < nValid) ? tanhf(tex[i]) : 0.0f;
}

// ---------------- 4) WMMA depth test / argmin over faces ----------------
// One wave handles 16 pixels of a row. Per 16-face tile:
//   D(16 faces x 16 pixels) = A(16x4 face coeffs) * B(4x16 pixel coords)
// 3 WMMAs per tile (w1, w2, invz). argmin(depth) == argmax(invz) among
// inside faces with invz > EPS; ties -> lowest face index.
__global__ void depth_kernel(const float4* __restrict__ P1,
                             const float4* __restrict__ P2,
                             const float4* __restrict__ PZ,
                             int* __restrict__ bestF,
                             int S, int Fp, int ntiles, int nGroups)
{
    const int lane = threadIdx.x & 31;
    const int wave = threadIdx.x >> 5;
    const int g = blockIdx.x * 8 + wave;
    if (g >= nGroups) return;
    const int gpr = S >> 4;                  // 16-pixel groups per row
    const int gpi = S * gpr;                 // groups per image
    const int b   = g / gpi;
    const int rem = g % gpi;
    const int row = rem / gpr;
    const int px0 = (rem % gpr) << 4;
    const int col16 = lane & 15;
    const int hsel  = lane >> 4;             // 0: lanes 0-15, 1: lanes 16-31
    const float Sf = (float)S;
    const float PX = (2.0f * (px0 + col16) + 1.0f - Sf) / Sf;
    const float PY = (2.0f * row + 1.0f - Sf) / Sf;

    // B operand (4x16, K x N): lanes 0-15 = (K0,K1) = (PX,PY); lanes 16-31 = (K2,K3) = (1,0)
    v2f Bm;
    Bm.x = hsel ? 1.0f : PX;
    Bm.y = hsel ? 0.0f : PY;

    const size_t fbase = (size_t)b * Fp;
    float best_iz = 0.0f;
    int   best_f  = -1;

    for (int t = 0; t < ntiles; ++t) {
        const size_t fo = fbase + (size_t)t * 16 + col16;
        // A operand (16x4, M x K): lanes 0-15 hold (a,b); lanes 16-31 hold (c,0)
        const float2* q1 = reinterpret_cast<const float2*>(P1 + fo);
        const float2* q2 = reinterpret_cast<const float2*>(P2 + fo);
        const float2* qz = reinterpret_cast<const float2*>(PZ + fo);
        float2 l1 = q1[hsel];
        float2 l2 = q2[hsel];
        float2 lz = qz[hsel];
        v2f A1; A1.x = l1.x; A1.y = l1.y;
        v2f A2; A2.x = l2.x; A2.y = l2.y;
        v2f AZ; AZ.x = lz.x; AZ.y = lz.y;
        v8f cz = {0.f, 0.f, 0.f, 0.f, 0.f, 0.f, 0.f, 0.f};
        v8f d1 = __builtin_amdgcn_wmma_f32_16x16x4_f32(false, A1, false, Bm, (short)0, cz, false, false);
        v8f d2 = __builtin_amdgcn_wmma_f32_16x16x4_f32(false, A2, false, Bm, (short)0, cz, false, false);
        v8f dz = __builtin_amdgcn_wmma_f32_16x16x4_f32(false, AZ, false, Bm, (short)0, cz, false, false);
        const int fb = t * 16 + hsel * 8;    // lane holds faces fb..fb+7 for pixel col16
#pragma unroll
        for (int r = 0; r < 8; ++r) {
            float w1 = d1[r], w2 = d2[r], iz = dz[r];
            float w0 = 1.0f - w1 - w2;
            bool inside = (w0 >= 0.0f) && (w1 >= 0.0f) && (w2 >= 0.0f) && (iz > EPSF);
            if (inside && iz > best_iz) { best_iz = iz; best_f = fb + r; }
        }
    }
    // combine lane halves (same pixel, complementary face subsets)
    float o_iz = __shfl_xor(best_iz, 16, 32);
    int   o_f  = __shfl_xor(best_f, 16, 32);
    bool take = (o_f >= 0) && (best_f < 0 || o_iz > best_iz ||
                               (o_iz == best_iz && o_f < best_f));
    if (take) { best_iz = o_iz; best_f = o_f; }
    if (lane < 16)
        bestF[((size_t)b * S + row) * S + px0 + col16] = best_f;
}

// ---------------- 5) shading + squared error, block partial sums ----------------
__global__ void shade_kernel(const int* __restrict__ bestF,
                             const float4* __restrict__ P1,
                             const float4* __restrict__ P2,
                             const float4* __restrict__ ZZ,
                             const float* __restrict__ TT,
                             const float* __restrict__ ref,
                             float* __restrict__ partials,
                             int S, int Fp, int total)
{
    int tid = blockIdx.x * blockDim.x + threadIdx.x;
    float acc = 0.0f;
    if (tid < total) {
        int b = tid / (S * S);
        int p = tid % (S * S);
        int y = p / S, x = p % S;
        int f = bestF[tid];
        float col0 = 0.f, col1 = 0.f, col2 = 0.f;
        if (f >= 0) {
            float Sf = (float)S;
            float PX = (2.0f * x + 1.0f - Sf) / Sf;
            float PY = (2.0f * y + 1.0f - Sf) / Sf;
            size_t o = (size_t)b * Fp + f;
            float4 c1 = P1[o], c2 = P2[o], zz = ZZ[o];
            float W1 = c1.x * PX + c1.y * PY + c1.z;
            float W2 = c2.x * PX + c2.y * PY + c2.z;
            float W0 = 1.0f - W1 - W2;
            float r0 = W0 / zz.x, r1 = W1 / zz.y, r2 = W2 / zz.z;
            float iz = fmaxf(r0 + r1 + r2, EPSF);
            float wp0 = fminf(fmaxf(r0 / iz, 0.0f), 1.0f);
            float wp1 = fminf(fmaxf(r1 / iz, 0.0f), 1.0f);
            float wp2 = fminf(fmaxf(r2 / iz, 0.0f), 1.0f);
            float A0[2] = {1.0f - wp0, wp0};
            float A1w[2] = {1.0f - wp1, wp1};
            float A2w[2] = {1.0f - wp2, wp2};
            const float* tf = TT + (size_t)f * 24;
#pragma unroll
            for (int i = 0; i < 2; ++i)
#pragma unroll
                for (int j = 0; j < 2; ++j)
#pragma unroll
                    for (int k = 0; k < 2; ++k) {
                        float w = A0[i] * A1w[j] * A2w[k];
                        const float* tc = tf + i * 12 + j * 6 + k * 3;
                        col0 += w * tc[0];
                        col1 += w * tc[1];
                        col2 += w * tc[2];
                    }
        }
        size_t rb = (size_t)b * 3 * S * S + (size_t)y * S + x;
        float d0 = col0 - ref[rb];
        float d1 = col1 - ref[rb + (size_t)S * S];
        float d2 = col2 - ref[rb + 2 * (size_t)S * S];
        acc = d0 * d0 + d1 * d1 + d2 * d2;
    }
    __shared__ float red[256];
    red[threadIdx.x] = acc;
    __syncthreads();
    for (int s = 128; s > 0; s >>= 1) {
        if (threadIdx.x < s) red[threadIdx.x] += red[threadIdx.x + s];
        __syncthreads();
    }
    if (threadIdx.x == 0) partials[blockIdx.x] = red[0];
}

// ---------------- 6) deterministic final reduction ----------------
__global__ void reduce_kernel(const float* __restrict__ partials, int n,
                              float* __restrict__ out)
{
    __shared__ float red[256];
    float a = 0.0f;
    for (int i = threadIdx.x; i < n; i += 256) a += partials[i];
    red[threadIdx.x] = a;
    __syncthreads();
    for (int s = 128; s > 0; s >>= 1) {
        if (threadIdx.x < s) red[threadIdx.x] += red[threadIdx.x + s];
        __syncthreads();
    }
    if (threadIdx.x == 0) out[0] = red[0];
}

extern "C" void kernel_launch(void* const* d_in, const int* in_sizes, int n_in,
                              void* d_out, int out_size, void* d_ws, size_t ws_size,
                              hipStream_t stream) {
    const float* verts = (const float*)d_in[0];
    const int*   faces = (const int*)d_in[1];
    const float* Km    = (const float*)d_in[2];
    const float* Rm    = (const float*)d_in[3];
    const float* tm    = (const float*)d_in[4];
    const float* tex   = (const float*)d_in[5];
    const float* ref   = (const float*)d_in[6];

    const int F = in_sizes[5] / 24;              // textures: (1,F,2,2,2,3)
    const int B = in_sizes[1] / (3 * F);         // faces: (B,F,3)
    const int V = in_sizes[0] / (3 * B);         // vertices: (B,V,3)
    const long pix = (long)in_sizes[6] / (3L * B);
    const int S = (int)llround(sqrt((double)pix));
    const int Fp = (F + 15) & ~15;

    // workspace carve (256B aligned slices)
    char* w = (char*)d_ws;
    size_t off = 0;
    auto carve = [&](size_t bytes) {
        void* p = w + off;
        off = (off + bytes + 255) & ~(size_t)255;
        return p;
    };
    float4* uvz = (float4*)carve((size_t)B * V * sizeof(float4));
    float4* P1  = (float4*)carve((size_t)B * Fp * sizeof(float4));
    float4* P2  = (float4*)carve((size_t)B * Fp * sizeof(float4));
    float4* PZ  = (float4*)carve((size_t)B * Fp * sizeof(float4));
    float4* ZZ  = (float4*)carve((size_t)B * Fp * sizeof(float4));
    float*  TT  = (float*)carve((size_t)Fp * 24 * sizeof(float));
    int*    bF  = (int*)carve((size_t)B * S * S * sizeof(int));
    const int total = B * S * S;
    const int shadeBlocks = (total + 255) / 256;
    float* partials = (float*)carve((size_t)shadeBlocks * sizeof(float));
    (void)ws_size; (void)n_in; (void)out_size;

    vert_kernel<<<(B * V + 255) / 256, 256, 0, stream>>>(verts, Km, Rm, tm, uvz, B, V, S);
    face_kernel<<<(B * Fp + 255) / 256, 256, 0, stream>>>(faces, uvz, P1, P2, PZ, ZZ, B, V, F, Fp);
    tanh_kernel<<<(Fp * 24 + 255) / 256, 256, 0, stream>>>(tex, TT, F * 24, Fp * 24);

    const int nGroups = B * S * (S >> 4);        // one wave per 16 pixels
    const int ntiles = Fp / 16;
    depth_kernel<<<(nGroups + 7) / 8, 256, 0, stream>>>(P1, P2, PZ, bF, S, Fp, ntiles, nGroups);

    shade_kernel<<<shadeBlocks, 256, 0, stream>>>(bF, P1, P2, ZZ, TT, ref, partials, S, Fp, total);
    reduce_kernel<<<1, 256, 0, stream>>>(partials, shadeBlocks, (float*)d_out);
}